// Qwen3Attention_87514253624094
// MI455X (gfx1250) — compile-verified
//
#include <hip/hip_runtime.h>
#include <hip/hip_bf16.h>
#include <math.h>

// ---------------------------------------------------------------------------
// Qwen3 decode-step attention for MI455X (gfx1250), fp32 end-to-end.
// HBM-bound problem (~420MB traffic, ~3.2 GFLOP) -> structure for bandwidth,
// use V_WMMA_F32_16X16X4_F32 for the projections (exact fp32 FMA semantics).
// ---------------------------------------------------------------------------

typedef float v2f __attribute__((ext_vector_type(2)));
typedef float v8f __attribute__((ext_vector_type(8)));

#define BB    32
#define HID   4096
#define NH    32
#define NHKV  8
#define DD    128
#define GG    4
#define MAXBLK 64
#define ATT_SCALE 0.088388347648318447f  // 1/sqrt(128)

// ---------------------------------------------------------------------------
// Generic GEMM: C[32 x N] = A[32 x 4096] * W[N x 4096]^T   (out[m][n]=sum_k A[m][k]W[n][k])
// One wave per 16-column tile; two 16x16 WMMA accumulators (M 0..15 and 16..31)
// sharing the B fragment. f32 WMMA layout (ISA 7.12.2):
//   A 16x4 : lane l<16 -> row M=l, K = {k,k+1};  lane>=16 -> row M=l-16, K = {k+2,k+3}
//   B 4x16 : lane l<16 -> col N=l, K = {k,k+1};  lane>=16 -> col N=l-16, K = {k+2,k+3}
//   C 16x16: vgpr r, lane l<16 -> (M=r,     N=l);  lane>=16 -> (M=r+8, N=l-16)
// ---------------------------------------------------------------------------
__global__ __launch_bounds__(32) void gemm_m32_wmma(
    const float* __restrict__ A, const float* __restrict__ W,
    float* __restrict__ C, int N) {
  const int K    = HID;
  const int lane = threadIdx.x & 31;
  const int col0 = blockIdx.x * 16;
  const int nl   = lane & 15;
  const int khalf = (lane >> 4) << 1;                // 0 or 2

  const float* a0p = A + (size_t)nl * K + khalf;     // rows 0..15
  const float* a1p = a0p + (size_t)16 * K;           // rows 16..31
  const float* wp  = W + (size_t)(col0 + nl) * K + khalf;

  v8f c0 = {0.f,0.f,0.f,0.f,0.f,0.f,0.f,0.f};
  v8f c1 = {0.f,0.f,0.f,0.f,0.f,0.f,0.f,0.f};

  for (int k0 = 0; k0 < K; k0 += 64) {
    __builtin_prefetch(wp + k0 + 1024, 0, 1);        // global_prefetch_b8 ahead in W
#pragma unroll
    for (int kk = 0; kk < 64; kk += 4) {
      v2f a0 = *(const v2f*)(a0p + k0 + kk);
      v2f a1 = *(const v2f*)(a1p + k0 + kk);
      v2f b  = *(const v2f*)(wp  + k0 + kk);
      c0 = __builtin_amdgcn_wmma_f32_16x16x4_f32(false, a0, false, b,
                                                 (short)0, c0, false, false);
      c1 = __builtin_amdgcn_wmma_f32_16x16x4_f32(false, a1, false, b,
                                                 (short)0, c1, false, false);
    }
  }

  const int mb = (lane >> 4) << 3;                   // 0 or 8
  const int nc = col0 + nl;
#pragma unroll
  for (int r = 0; r < 8; ++r) {
    C[(size_t)(mb + r)      * N + nc] = c0[r];
    C[(size_t)(mb + r + 16) * N + nc] = c1[r];
  }
}

// ---------------------------------------------------------------------------
// Per-head RMSNorm + RoPE, in place on the projected q / k buffers.
// Block = 128 threads = one (token, head). Blocks [0, B*H) -> q, rest -> k.
// ---------------------------------------------------------------------------
__global__ __launch_bounds__(128) void normrope_kernel(
    float* __restrict__ qbuf, float* __restrict__ kbuf,
    const float* __restrict__ qw, const float* __restrict__ kw,
    const float* __restrict__ cosv, const float* __restrict__ sinv) {
  __shared__ float red[128];
  __shared__ float xs[128];
  const int t   = threadIdx.x;
  const int idx = blockIdx.x;

  float* buf; const float* w; int b;
  if (idx < BB * NH) {
    b = idx >> 5; buf = qbuf + (size_t)idx * DD; w = qw;
  } else {
    int i2 = idx - BB * NH;
    b = i2 >> 3; buf = kbuf + (size_t)i2 * DD; w = kw;
  }

  float x = buf[t];
  red[t] = x * x;
  __syncthreads();
#pragma unroll
  for (int s = 64; s > 0; s >>= 1) {
    if (t < s) red[t] += red[t + s];
    __syncthreads();
  }
  const float inv = rsqrtf(red[0] * (1.0f / DD) + 1e-6f);
  const float xn  = x * inv * w[t];
  xs[t] = xn;
  __syncthreads();

  const float part = xs[(t + 64) & 127];
  const float cv   = cosv[b * DD + t];
  const float sv   = sinv[b * DD + t];
  const float rot  = (t < 64) ? -part : part;       // rotate_half
  buf[t] = xn * cv + rot * sv;
}

// ---------------------------------------------------------------------------
// Flash-decode GQA attention. Block = 256 threads (8 waves) per (b, kv head).
// Each wave streams keys l = w, w+8, ... with coalesced float4 loads
// (one 512B key row per wave), online softmax for the 4 query heads of the
// group, wave-ballot lookup of slot_mapping so new k/v come from workspace
// (inputs are never mutated), LDS merge of the 8 partial softmaxes.
// ---------------------------------------------------------------------------
__global__ __launch_bounds__(256) void attn_decode_kernel(
    const float* __restrict__ qbuf,   // [B][H][D]    (post norm+rope)
    const float* __restrict__ kbuf,   // [B][HKV][D]  (post norm+rope)
    const float* __restrict__ vbuf,   // [B][HKV][D]
    const float* __restrict__ kc,     // [NB*16][HKV][D]
    const float* __restrict__ vc,
    const int* __restrict__ bt,       // [B][64]
    const int* __restrict__ ctx_lens, // [B]
    const int* __restrict__ slotmap,  // [B]
    float* __restrict__ out) {        // [B][H*D]
  __shared__ float sh_m[8][GG];
  __shared__ float sh_s[8][GG];
  __shared__ float sh_acc[8][GG][DD];

  const int bh   = blockIdx.x;
  const int b    = bh >> 3;
  const int h    = bh & 7;
  const int tid  = threadIdx.x;
  const int w    = tid >> 5;
  const int lane = tid & 31;

  const int myslot = slotmap[lane];          // one mapped slot per lane
  const int ctx    = ctx_lens[b];

  float4 q4[GG];
#pragma unroll
  for (int g = 0; g < GG; ++g)
    q4[g] = *(const float4*)(qbuf + ((size_t)(b * NH + h * GG + g)) * DD + 4 * lane);

  float  m[GG], ssum[GG];
  float4 acc[GG];
#pragma unroll
  for (int g = 0; g < GG; ++g) {
    m[g] = -1.0e30f; ssum[g] = 0.f;
    acc[g] = make_float4(0.f, 0.f, 0.f, 0.f);
  }

  for (int l = w; l < ctx; l += 8) {
    int blk = bt[b * MAXBLK + (l >> 4)];
    if (blk < 0) blk = 0;                    // jnp.clip(block_tables, 0)
    const int slot = blk * 16 + (l & 15);

    // was this slot just overwritten by this batch's scatter? (wave-uniform)
    unsigned long long mask = __ballot(myslot == slot);
    const float *kp, *vp;
    if (mask) {
      const int j = __ffsll((unsigned long long)mask) - 1;
      kp = kbuf + (size_t)j * (NHKV * DD) + h * DD;
      vp = vbuf + (size_t)j * (NHKV * DD) + h * DD;
    } else {
      kp = kc + (size_t)slot * (NHKV * DD) + h * DD;
      vp = vc + (size_t)slot * (NHKV * DD) + h * DD;
    }
    const float4 kf = *(const float4*)(kp + 4 * lane);
    const float4 vf = *(const float4*)(vp + 4 * lane);

#pragma unroll
    for (int g = 0; g < GG; ++g) {
      float p = q4[g].x * kf.x + q4[g].y * kf.y + q4[g].z * kf.z + q4[g].w * kf.w;
#pragma unroll
      for (int off = 16; off > 0; off >>= 1) p += __shfl_xor(p, off, 32);
      p *= ATT_SCALE;

      const float mn   = fmaxf(m[g], p);
      const float corr = __expf(m[g] - mn);
      const float e    = __expf(p - mn);
      ssum[g] = ssum[g] * corr + e;
      acc[g].x = acc[g].x * corr + e * vf.x;
      acc[g].y = acc[g].y * corr + e * vf.y;
      acc[g].z = acc[g].z * corr + e * vf.z;
      acc[g].w = acc[g].w * corr + e * vf.w;
      m[g] = mn;
    }
  }

#pragma unroll
  for (int g = 0; g < GG; ++g) {
    if (lane == 0) { sh_m[w][g] = m[g]; sh_s[w][g] = ssum[g]; }
    sh_acc[w][g][4 * lane + 0] = acc[g].x;
    sh_acc[w][g][4 * lane + 1] = acc[g].y;
    sh_acc[w][g][4 * lane + 2] = acc[g].z;
    sh_acc[w][g][4 * lane + 3] = acc[g].w;
  }
  __syncthreads();

  for (int idx = tid; idx < GG * DD; idx += 256) {
    const int g = idx >> 7;
    const int d = idx & 127;
    float M = -1.0e30f;
#pragma unroll
    for (int ww = 0; ww < 8; ++ww) M = fmaxf(M, sh_m[ww][g]);
    float tot = 0.f, o = 0.f;
#pragma unroll
    for (int ww = 0; ww < 8; ++ww) {
      if (sh_s[ww][g] > 0.f) {
        const float f = __expf(sh_m[ww][g] - M);
        tot += sh_s[ww][g] * f;
        o   += sh_acc[ww][g][d] * f;
      }
    }
    out[(size_t)b * (NH * DD) + (h * GG + g) * DD + d] = o / tot;
  }
}

// ---------------------------------------------------------------------------
extern "C" void kernel_launch(void* const* d_in, const int* in_sizes, int n_in,
                              void* d_out, int out_size, void* d_ws, size_t ws_size,
                              hipStream_t stream) {
  const float* hs   = (const float*)d_in[0];
  const float* cosv = (const float*)d_in[1];
  const float* sinv = (const float*)d_in[2];
  const float* kc   = (const float*)d_in[3];
  const float* vc   = (const float*)d_in[4];
  const float* Wq   = (const float*)d_in[5];
  const float* Wk   = (const float*)d_in[6];
  const float* Wv   = (const float*)d_in[7];
  const float* Wo   = (const float*)d_in[8];
  const float* qw   = (const float*)d_in[9];
  const float* kw   = (const float*)d_in[10];
  const int*   bt   = (const int*)d_in[11];
  const int*   cl   = (const int*)d_in[12];
  const int*   sm   = (const int*)d_in[13];
  float*       out  = (float*)d_out;

  float* ws_q    = (float*)d_ws;                 // B * 4096
  float* ws_k    = ws_q + BB * HID;              // B * 1024
  float* ws_v    = ws_k + BB * NHKV * DD;        // B * 1024
  float* ws_attn = ws_v + BB * NHKV * DD;        // B * 4096

  // 1) fused-precision-exact QKV projections (WMMA f32)
  gemm_m32_wmma<<<(NH * DD) / 16, 32, 0, stream>>>(hs, Wq, ws_q, NH * DD);
  gemm_m32_wmma<<<(NHKV * DD) / 16, 32, 0, stream>>>(hs, Wk, ws_k, NHKV * DD);
  gemm_m32_wmma<<<(NHKV * DD) / 16, 32, 0, stream>>>(hs, Wv, ws_v, NHKV * DD);

  // 2) per-head RMSNorm + RoPE (q then k sections)
  normrope_kernel<<<BB * NH + BB * NHKV, 128, 0, stream>>>(
      ws_q, ws_k, qw, kw, cosv, sinv);

  // 3) paged flash-decode attention with slot-mapping patch (no input mutation)
  attn_decode_kernel<<<BB * NHKV, 256, 0, stream>>>(
      ws_q, ws_k, ws_v, kc, vc, bt, cl, sm, ws_attn);

  // 4) output projection (WMMA f32) straight into d_out
  gemm_m32_wmma<<<HID / 16, 32, 0, stream>>>(ws_attn, Wo, out, HID);
}